// GCNLayer_53609781789055
// MI455X (gfx1250) — compile-verified
//
#include <hip/hip_runtime.h>
#include <cstddef>

#define N_NODES 100000
#define N_EDGES 1600000
#define D 64
#define BN_EPS 1e-5f

typedef __attribute__((ext_vector_type(2))) float v2f;
typedef __attribute__((ext_vector_type(8))) float v8f;

// ---------------------------------------------------------------------------
// Kernel 1: zero agg + stats scratch
// ---------------------------------------------------------------------------
__global__ __launch_bounds__(256) void gcn_zero(float4* __restrict__ p, int n4) {
    int i = blockIdx.x * 256 + threadIdx.x;
    if (i < n4) p[i] = make_float4(0.f, 0.f, 0.f, 0.f);
}

// ---------------------------------------------------------------------------
// Kernel 2: support = x @ W via V_WMMA_F32_16X16X4_F32 (wave32, 16x16 tiles)
// Each wave: 16 rows of x, all 64 output cols (4 N-tiles), K=64 in 16 steps.
// A frag (16x4 f32): lanes 0-15 -> (M=lane, K=2*half+{0,1})  -> float2 load.
// B frag (4x16 f32): lane l -> (K=2*half+{0,1}, N=ntile*16+l) from LDS-staged W.
// D (16x16 f32): VGPR j -> (M=row_base+j+8*half, N=ntile*16+l).
// ---------------------------------------------------------------------------
__global__ __launch_bounds__(256) void gcn_gemm_wmma(
    const float* __restrict__ x, const float* __restrict__ W,
    float* __restrict__ support)
{
    __shared__ float Wl[D * D];                 // 16 KB of the 320 KB WGP LDS
    {
        const float4* Wv  = (const float4*)W;
        float4*       Wlv = (float4*)Wl;
        for (int i = threadIdx.x; i < D * D / 4; i += 256) Wlv[i] = Wv[i];
    }
    __syncthreads();

    const int wave = threadIdx.x >> 5;
    const int lane = threadIdx.x & 31;
    const int half = lane >> 4;                 // 0: lanes 0-15, 1: lanes 16-31
    const int l    = lane & 15;

    const int row_base = (blockIdx.x * 8 + wave) * 16;
    if (row_base >= N_NODES) return;            // wave-uniform; EXEC stays all-1s

    const float* xrow = x + (size_t)(row_base + l) * D;

    v8f acc[4] = {};

#pragma unroll
    for (int k = 0; k < D / 4; ++k) {
        const int krow = k * 4 + 2 * half;
        v2f a;
        a.x = xrow[krow];
        a.y = xrow[krow + 1];
#pragma unroll
        for (int n = 0; n < 4; ++n) {
            v2f b;
            b.x = Wl[krow * D + n * 16 + l];
            b.y = Wl[(krow + 1) * D + n * 16 + l];
            acc[n] = __builtin_amdgcn_wmma_f32_16x16x4_f32(
                /*neg_a=*/false, a, /*neg_b=*/false, b,
                /*c_mod=*/(short)0, acc[n], /*reuse_a=*/false, /*reuse_b=*/false);
        }
    }

#pragma unroll
    for (int n = 0; n < 4; ++n) {
#pragma unroll
        for (int j = 0; j < 8; ++j) {
            support[(size_t)(row_base + j + 8 * half) * D + n * 16 + l] = acc[n][j];
        }
    }
}

// ---------------------------------------------------------------------------
// Kernel 3: COO scatter-add: agg[dst] += w * support[src]
// 16 threads per edge, float4 gather (L2-resident: support+agg = 51MB << 192MB L2),
// native fp32 global atomics (GLOBAL_ATOMIC_ADD_F32).
// ---------------------------------------------------------------------------
__global__ __launch_bounds__(256) void gcn_scatter(
    const float* __restrict__ support, const int* __restrict__ esrc,
    const int* __restrict__ edst, const float* __restrict__ ew,
    float* __restrict__ agg)
{
    const int gid = blockIdx.x * 256 + threadIdx.x;     // N_EDGES*16 threads
    const int e = gid >> 4;
    if (e >= N_EDGES) return;
    const int c = (gid & 15) * 4;

    const int   s = esrc[e];
    const int   d = edst[e];
    const float w = ew[e];

    const float4 v = *(const float4*)(support + (size_t)s * D + c);
    float* ap = agg + (size_t)d * D + c;
    __hip_atomic_fetch_add(ap + 0, w * v.x, __ATOMIC_RELAXED, __HIP_MEMORY_SCOPE_AGENT);
    __hip_atomic_fetch_add(ap + 1, w * v.y, __ATOMIC_RELAXED, __HIP_MEMORY_SCOPE_AGENT);
    __hip_atomic_fetch_add(ap + 2, w * v.z, __ATOMIC_RELAXED, __HIP_MEMORY_SCOPE_AGENT);
    __hip_atomic_fetch_add(ap + 3, w * v.w, __ATOMIC_RELAXED, __HIP_MEMORY_SCOPE_AGENT);
}

// ---------------------------------------------------------------------------
// Kernel 4: per-column sum / sumsq of (agg + bias), LDS tree + global atomics
// ---------------------------------------------------------------------------
#define STATS_ROWS 512
__global__ __launch_bounds__(256) void gcn_stats(
    const float* __restrict__ agg, const float* __restrict__ bias,
    float* __restrict__ sums)   // sums[0..63]=sum, sums[64..127]=sumsq
{
    __shared__ float ls[256];
    __shared__ float lss[256];
    const int c = threadIdx.x & 63;
    const int r = threadIdx.x >> 6;   // 0..3

    const float b = bias[c];
    float s = 0.f, ss = 0.f;
    const int row0 = blockIdx.x * STATS_ROWS;
    const int row1 = min(row0 + STATS_ROWS, N_NODES);
    for (int row = row0 + r; row < row1; row += 4) {
        const float v = agg[(size_t)row * D + c] + b;
        s  += v;
        ss += v * v;
    }
    ls[threadIdx.x]  = s;
    lss[threadIdx.x] = ss;
    __syncthreads();
    if (r == 0) {
        s  = ls[c]  + ls[64 + c]  + ls[128 + c]  + ls[192 + c];
        ss = lss[c] + lss[64 + c] + lss[128 + c] + lss[192 + c];
        __hip_atomic_fetch_add(&sums[c],      s,  __ATOMIC_RELAXED, __HIP_MEMORY_SCOPE_AGENT);
        __hip_atomic_fetch_add(&sums[64 + c], ss, __ATOMIC_RELAXED, __HIP_MEMORY_SCOPE_AGENT);
    }
}

// ---------------------------------------------------------------------------
// Kernel 5: fold bias/mean/var/gamma/beta into per-column scale & shift
// out = relu((agg + bias - mean) * gamma * rsqrt(var+eps) + beta)
//     = relu(agg * scale + shift),  shift = beta + (bias - mean) * scale
// ---------------------------------------------------------------------------
__global__ void gcn_bnparams(float* __restrict__ sums,
                             const float* __restrict__ bias,
                             const float* __restrict__ gamma,
                             const float* __restrict__ beta)
{
    const int c = threadIdx.x;              // 64 threads
    const float inv_n = 1.0f / (float)N_NODES;
    const float mean  = sums[c] * inv_n;
    const float var   = sums[64 + c] * inv_n - mean * mean;
    const float scale = gamma[c] * rsqrtf(var + BN_EPS);
    sums[c]      = scale;
    sums[64 + c] = beta[c] + (bias[c] - mean) * scale;
}

// ---------------------------------------------------------------------------
// Kernel 6: fused BN + ReLU, float4-vectorized single pass
// ---------------------------------------------------------------------------
__global__ __launch_bounds__(256) void gcn_bnrelu(
    const float* __restrict__ agg, const float* __restrict__ sums,
    float* __restrict__ out)
{
    const int gid = blockIdx.x * 256 + threadIdx.x;     // N_NODES*16 threads
    if (gid >= N_NODES * 16) return;
    const int c = (gid & 15) * 4;
    const size_t off = (size_t)(gid >> 4) * D + c;

    const float4 v     = *(const float4*)(agg + off);
    const float4 scale = *(const float4*)(sums + c);
    const float4 shift = *(const float4*)(sums + 64 + c);

    float4 o;
    o.x = fmaxf(0.f, fmaf(v.x, scale.x, shift.x));
    o.y = fmaxf(0.f, fmaf(v.y, scale.y, shift.y));
    o.z = fmaxf(0.f, fmaf(v.z, scale.z, shift.z));
    o.w = fmaxf(0.f, fmaf(v.w, scale.w, shift.w));
    *(float4*)(out + off) = o;
}

// ---------------------------------------------------------------------------
extern "C" void kernel_launch(void* const* d_in, const int* in_sizes, int n_in,
                              void* d_out, int out_size, void* d_ws, size_t ws_size,
                              hipStream_t stream) {
    const float* x     = (const float*)d_in[0];
    const int*   esrc  = (const int*)  d_in[1];
    const int*   edst  = (const int*)  d_in[2];
    const float* ew    = (const float*)d_in[3];
    const float* W     = (const float*)d_in[4];
    const float* bias  = (const float*)d_in[5];
    const float* gamma = (const float*)d_in[6];
    const float* beta  = (const float*)d_in[7];

    float* out     = (float*)d_out;
    float* agg     = (float*)d_ws;                       // N*64 floats (25.6 MB)
    float* sums    = agg + (size_t)N_NODES * D;          // 128 floats
    float* support = out;                                // reuse d_out as scratch

    // 1) zero agg + stats (ws is poisoned, must init)
    const int n4 = N_NODES * D / 4 + 32;
    gcn_zero<<<dim3((n4 + 255) / 256), dim3(256), 0, stream>>>((float4*)agg, n4);

    // 2) support = x @ W  (WMMA)
    gcn_gemm_wmma<<<dim3((N_NODES / 16 + 7) / 8), dim3(256), 0, stream>>>(x, W, support);

    // 3) agg[dst] += w * support[src]
    gcn_scatter<<<dim3(N_EDGES * 16 / 256), dim3(256), 0, stream>>>(
        support, esrc, edst, ew, agg);

    // 4) column stats
    gcn_stats<<<dim3((N_NODES + STATS_ROWS - 1) / STATS_ROWS), dim3(256), 0, stream>>>(
        agg, bias, sums);

    // 5) fold into scale/shift
    gcn_bnparams<<<dim3(1), dim3(64), 0, stream>>>(sums, bias, gamma, beta);

    // 6) fused BN + ReLU -> d_out
    gcn_bnrelu<<<dim3(N_NODES * 16 / 256), dim3(256), 0, stream>>>(agg, sums, out);
}